// GCN_15418932593106
// MI455X (gfx1250) — compile-verified
//
#include <hip/hip_runtime.h>

// GCNConv(1,1,normalize=False,bias=False) + take every 3rd node:
//   out[i] = sum_{edges (s->d), d==3*i} x[s] * w
//
// N_NODES = 3,000,000  N_EDGES = 48,000,000
// d_in[0]=x (f32,3M)  d_in[1]=edge_index (i64, src[48M] then dst[48M])  d_in[2]=W (f32,1)
// d_out = 1M f32
//
// MI455X plan:
//  - 768 MB single-use edge stream: async global->LDS copy engine (ASYNCcnt),
//    double-buffered per wave, NT hint so it doesn't thrash the 192 MB L2.
//  - x (12 MB) + out (4 MB) stay L2-resident; only dst%3==0 edges (~16M/48M)
//    gather x and issue global_atomic_add_f32.

typedef __attribute__((ext_vector_type(2))) long long v2i64;
typedef __attribute__((address_space(3))) char LdsChar;
typedef __attribute__((address_space(3))) const v2i64 LdsV2;

#define GCN_STR1(x) #x
#define GCN_STR(x) GCN_STR1(x)

// One async b128 copy per lane: copies 16 bytes from global memory at saddr + voff + OFF into LDS at lds32 + OFF
// (ISA: INST_OFFSET is added to BOTH the LDS and global addresses.)
#define GCN_ASYNC_B128(lds32, voff, saddr, OFF)                                   \
    asm volatile("global_load_async_to_lds_b128 %0, %1, %2 offset:" GCN_STR(OFF)  \
                 " th:TH_LOAD_NT"                                                 \
                 :: "v"(lds32), "v"(voff), "s"(saddr) : "memory")

// Tile geometry: 256 edges per stream per wave-tile.
//   bytes/stream/tile = 2048; per lane: 16 B x 4 chunks (chunk stride 512 B).
#define TILE_EDGES   256
#define TILE_BYTES   2048           // per stream
#define BUF_BYTES    4096           // src(2KB) + dst(2KB)
#define WAVE_BYTES   8192           // double buffer
#define WAVES_PER_BLOCK 8

__global__ __launch_bounds__(256) void GCN_zero_kernel(float* __restrict__ out, int n) {
    int i = blockIdx.x * blockDim.x + threadIdx.x;
    int stride = gridDim.x * blockDim.x;
    for (; i < n; i += stride) out[i] = 0.0f;
}

__device__ __forceinline__ void gcn_edge(unsigned s, unsigned d,
                                         const float* __restrict__ x, float w,
                                         float* __restrict__ out) {
    if (d % 3u == 0u) {
        atomicAdd(out + (d / 3u), x[s] * w);   // global_atomic_add_f32 (no return)
    }
}

__global__ __launch_bounds__(256) void GCN_scatter_kernel(
    const long long* __restrict__ src,
    const long long* __restrict__ dst,
    const float* __restrict__ x,
    const float* __restrict__ Wp,
    float* __restrict__ out,
    long long n_edges)
{
    __shared__ char smem[WAVES_PER_BLOCK * WAVE_BYTES];   // 64 KB / block

    const float w = Wp[0];

    const int lane = threadIdx.x & 31;
    const int wave = threadIdx.x >> 5;

    LdsChar* lwave = (LdsChar*)smem + wave * WAVE_BYTES;
    const unsigned lds_wave = (unsigned)(size_t)lwave;

    const long long gw = (long long)blockIdx.x * WAVES_PER_BLOCK + wave;  // wave id
    const long long GW = (long long)gridDim.x * WAVES_PER_BLOCK;          // total waves
    const long long n_tiles = n_edges / TILE_EDGES;

    const unsigned laneB = (unsigned)lane * 16u;

    // ---- issue one tile's async copies into buffer b (8 x b128 per lane) ----
    auto issue = [&](long long t, unsigned b) {
        // WAR guard: our own ds reads of this buffer (previous iteration) may still
        // be pending if a predicated use was branched over.
        asm volatile("s_wait_dscnt 0x0" ::: "memory");
        const unsigned lsrc = lds_wave + b * BUF_BYTES + laneB;
        const unsigned ldst = lsrc + TILE_BYTES;
        const unsigned voff = (unsigned)(t * TILE_BYTES) + laneB;
        GCN_ASYNC_B128(lsrc, voff, src, 0);
        GCN_ASYNC_B128(lsrc, voff, src, 512);
        GCN_ASYNC_B128(lsrc, voff, src, 1024);
        GCN_ASYNC_B128(lsrc, voff, src, 1536);
        GCN_ASYNC_B128(ldst, voff, dst, 0);
        GCN_ASYNC_B128(ldst, voff, dst, 512);
        GCN_ASYNC_B128(ldst, voff, dst, 1024);
        GCN_ASYNC_B128(ldst, voff, dst, 1536);
    };

    // ---- consume one tile from buffer b: conflict-free lane*16-strided reads ----
    auto process = [&](unsigned b) {
        LdsChar* ps = lwave + b * BUF_BYTES + laneB;
        LdsChar* pd = ps + TILE_BYTES;
#pragma unroll
        for (int k = 0; k < 4; ++k) {
            v2i64 s2 = *(LdsV2*)(ps + k * 512);   // ds_load_b128
            v2i64 d2 = *(LdsV2*)(pd + k * 512);   // ds_load_b128
            gcn_edge((unsigned)s2.x, (unsigned)d2.x, x, w, out);
            gcn_edge((unsigned)s2.y, (unsigned)d2.y, x, w, out);
        }
    };

    // ---- software-pipelined double buffer over this wave's tiles ----
    long long t = gw;
    unsigned buf = 0;
    if (t < n_tiles) issue(t, 0);
    while (t < n_tiles) {
        const long long tn = t + GW;
        if (tn < n_tiles) {
            issue(tn, buf ^ 1u);
            // previous buffer done when only the 8 just-issued ops remain
            asm volatile("s_wait_asynccnt 0x8" ::: "memory");
        } else {
            asm volatile("s_wait_asynccnt 0x0" ::: "memory");
        }
        process(buf);
        buf ^= 1u;
        t = tn;
    }

    // ---- tail edges (n_edges % 256), direct loads ----
    const long long tail = n_tiles * TILE_EDGES;
    const long long tid  = (long long)blockIdx.x * blockDim.x + threadIdx.x;
    const long long nthr = (long long)gridDim.x * blockDim.x;
    for (long long e = tail + tid; e < n_edges; e += nthr) {
        gcn_edge((unsigned)src[e], (unsigned)dst[e], x, w, out);
    }
}

extern "C" void kernel_launch(void* const* d_in, const int* in_sizes, int n_in,
                              void* d_out, int out_size, void* d_ws, size_t ws_size,
                              hipStream_t stream) {
    const float*     x     = (const float*)d_in[0];
    const long long* edges = (const long long*)d_in[1];
    const float*     W     = (const float*)d_in[2];
    float*           out   = (float*)d_out;

    const long long n_edges = (long long)in_sizes[1] / 2;   // 48M
    const long long* src = edges;
    const long long* dst = edges + n_edges;

    // 1) zero the (poisoned) output accumulator
    {
        int threads = 256;
        int blocks  = (out_size + threads - 1) / threads;
        if (blocks > 2048) blocks = 2048;
        if (blocks < 1) blocks = 1;
        GCN_zero_kernel<<<blocks, threads, 0, stream>>>(out, out_size);
    }

    // 2) async-pipelined edge scatter-add
    {
        const long long n_tiles = n_edges / TILE_EDGES;     // 187,500
        long long want_blocks = (n_tiles + WAVES_PER_BLOCK - 1) / WAVES_PER_BLOCK;
        int blocks = (want_blocks > 2048) ? 2048 : (int)want_blocks;
        if (blocks < 1) blocks = 1;
        GCN_scatter_kernel<<<blocks, 256, 0, stream>>>(src, dst, x, W, out, n_edges);
    }
}